// GCN_18159121727818
// MI455X (gfx1250) — compile-verified
//
#include <hip/hip_runtime.h>
#include <math.h>

// ---------------------------------------------------------------------------
// GCN + FISTA dictionary-learning pipeline for MI455X (gfx1250, wave32, WMMA)
//
// Heavy GEMMs run through V_WMMA_F32_16X16X4_F32 (exact fp32 matrix path).
// Block tile 128x128, 8 waves (2x4), 64x32 per wave -> 8 WMMA per 6 fragment
// loads per k-step (vs 4:4 before). global_prefetch_b8 of next K-chunk.
// ---------------------------------------------------------------------------

typedef __attribute__((ext_vector_type(2))) float v2f;
typedef __attribute__((ext_vector_type(8))) float v8f;

// Only check builtin availability in the DEVICE pass: __has_builtin for
// amdgcn builtins is 0 in the host (x86) pass of HIP's dual compilation.
#if defined(__HIP_DEVICE_COMPILE__)
#if !__has_builtin(__builtin_amdgcn_wmma_f32_16x16x4_f32)
#error "missing __builtin_amdgcn_wmma_f32_16x16x4_f32 on device toolchain"
#endif
#endif

static __device__ __forceinline__ v8f wmma4(v2f a, v2f b, v8f c) {
  // (neg_a, A, neg_b, B, c_mod, C, reuse_a, reuse_b)
  return __builtin_amdgcn_wmma_f32_16x16x4_f32(false, a, false, b, (short)0, c,
                                               false, false);
}

static __device__ __forceinline__ float softthr(float x, float t) {
  float a = fabsf(x) - t;
  a = a > 0.0f ? a : 0.0f;
  return copysignf(a, x);  // sign(x) * relu(|x| - t)
}

#define BM 128
#define BN 128
#define BK 32
#define LDS_STR 34  // padded row stride (floats): even -> b64-aligned, bank-spread

// EPI: 0 plain C=AB
//      1 residual: C = AB - Yb (elementwise), sum-of-squares atomicAdd -> ssq
//      2 Gamma0 : g = softthr(eta*AB, lam);           Gbuf=g, Zbuf=g
//      3 FISTA  : g = softthr(Zbuf - eta*AB, lam/c);  Zbuf=g+mom*(g-Gold); Gbuf=g
//      4 bias+relu: C = relu(AB + Yb[col])
//      5 bias     : C = AB + Yb[col]
template <int TRANSA, int EPI>
__global__ __launch_bounds__(256) void gemm_wmma(
    int M, int N, int K, const float* __restrict__ A, int lda,
    const float* __restrict__ B, int ldb, float* __restrict__ C, int ldc,
    const float* __restrict__ Yb, float* __restrict__ Gbuf,
    float* __restrict__ Zbuf, const float* __restrict__ cptr,
    const int* __restrict__ lamptr, float mom, float* __restrict__ ssq) {
  __shared__ __align__(16) float As[BM * LDS_STR];
  __shared__ __align__(16) float Bs[BN * LDS_STR];
  __shared__ float red[256];

  const int tid = threadIdx.x;
  const int m0 = blockIdx.y * BM;
  const int n0 = blockIdx.x * BN;
  const int wave = tid >> 5;
  const int lane = tid & 31;
  const int wm = wave >> 2;  // 0..1 -> 64-row slab
  const int wn = wave & 3;   // 0..3 -> 32-col slab
  const int lr = lane & 15;
  const int hi = lane >> 4;

  v8f acc[4][2];
#pragma unroll
  for (int i = 0; i < 4; ++i)
#pragma unroll
    for (int j = 0; j < 2; ++j)
#pragma unroll
      for (int v = 0; v < 8; ++v) acc[i][j][v] = 0.0f;

  const int nkb = (K + BK - 1) / BK;
  for (int kb = 0; kb < nkb; ++kb) {
    const int k0 = kb * BK;
    // ---- stage A tile: As[m][k], m-major, k contiguous
    if (TRANSA == 0) {
#pragma unroll
      for (int i = 0; i < 16; ++i) {
        int idx = tid + i * 256;      // 128*32 = 4096 elements
        int r = idx >> 5;             // m in tile
        int kk = idx & 31;            // k in tile (coalesced across lanes)
        int gm = m0 + r, gk = k0 + kk;
        float v = (gm < M && gk < K) ? A[(long long)gm * lda + gk] : 0.0f;
        As[r * LDS_STR + kk] = v;
      }
    } else {
      // logical A[m][k] = Aptr[k*lda + m]; coalesce along m
#pragma unroll
      for (int i = 0; i < 16; ++i) {
        int idx = tid + i * 256;
        int r = idx & 127;            // m (coalesced)
        int kk = idx >> 7;            // k
        int gm = m0 + r, gk = k0 + kk;
        float v = (gm < M && gk < K) ? A[(long long)gk * lda + gm] : 0.0f;
        As[r * LDS_STR + kk] = v;
      }
    }
    // ---- stage B tile transposed: Bs[n][k], n-major, k contiguous
#pragma unroll
    for (int i = 0; i < 16; ++i) {
      int idx = tid + i * 256;        // 128*32 = 4096 elements
      int n = idx & 127;              // n (coalesced)
      int kk = idx >> 7;              // k
      int gn = n0 + n, gk = k0 + kk;
      float v = (gn < N && gk < K) ? B[(long long)gk * ldb + gn] : 0.0f;
      Bs[n * LDS_STR + kk] = v;
    }
    // ---- prefetch next K-chunk into cache (global_prefetch_b8),
    //      bounds-clamped so the address is always valid.
    if (kb + 1 < nkb) {
      int pk = k0 + BK + (tid & 31);
      pk = pk < K ? pk : K - 1;
      int pa = m0 + (tid >> 5) * 16;
      pa = pa < M ? pa : M - 1;
      int pb = n0 + (tid >> 5) * 16;
      pb = pb < N ? pb : N - 1;
      if (TRANSA == 0)
        __builtin_prefetch(&A[(long long)pa * lda + pk], 0, 3);
      else
        __builtin_prefetch(&A[(long long)pk * lda + pa], 0, 3);
      __builtin_prefetch(&B[(long long)pk * ldb + pb], 0, 3);
    }
    __syncthreads();

#pragma unroll
    for (int ks = 0; ks < 8; ++ks) {
      // fp32 16x16x4 fragment layout: lanes 0-15 hold K0,K1; lanes 16-31 K2,K3
      int kk = ks * 4 + hi * 2;
      v2f a0 = *reinterpret_cast<const v2f*>(&As[(wm * 64 + lr) * LDS_STR + kk]);
      v2f a1 =
          *reinterpret_cast<const v2f*>(&As[(wm * 64 + 16 + lr) * LDS_STR + kk]);
      v2f a2 =
          *reinterpret_cast<const v2f*>(&As[(wm * 64 + 32 + lr) * LDS_STR + kk]);
      v2f a3 =
          *reinterpret_cast<const v2f*>(&As[(wm * 64 + 48 + lr) * LDS_STR + kk]);
      v2f b0 = *reinterpret_cast<const v2f*>(&Bs[(wn * 32 + lr) * LDS_STR + kk]);
      v2f b1 =
          *reinterpret_cast<const v2f*>(&Bs[(wn * 32 + 16 + lr) * LDS_STR + kk]);
      acc[0][0] = wmma4(a0, b0, acc[0][0]);
      acc[0][1] = wmma4(a0, b1, acc[0][1]);
      acc[1][0] = wmma4(a1, b0, acc[1][0]);
      acc[1][1] = wmma4(a1, b1, acc[1][1]);
      acc[2][0] = wmma4(a2, b0, acc[2][0]);
      acc[2][1] = wmma4(a2, b1, acc[2][1]);
      acc[3][0] = wmma4(a3, b0, acc[3][0]);
      acc[3][1] = wmma4(a3, b1, acc[3][1]);
    }
    __syncthreads();
  }

  // ---- epilogue
  float eta = 0.0f, thr = 0.0f;
  if (EPI == 2 || EPI == 3) {
    float c = cptr[0];
    float lam = (float)lamptr[0];
    eta = 1.0f / c;
    thr = (EPI == 2) ? lam : lam / c;
  }
  float lsum = 0.0f;
#pragma unroll
  for (int i = 0; i < 4; ++i)
#pragma unroll
    for (int j = 0; j < 2; ++j)
#pragma unroll
      for (int v = 0; v < 8; ++v) {
        // D layout: VGPR v holds M=v (lanes 0-15) / M=v+8 (lanes 16-31)
        int r = m0 + wm * 64 + i * 16 + hi * 8 + v;
        int cl = n0 + wn * 32 + j * 16 + lr;
        if (r < M && cl < N) {
          float val = acc[i][j][v];
          long long o = (long long)r * ldc + cl;
          if (EPI == 0) {
            C[o] = val;
          } else if (EPI == 1) {
            float d = val - Yb[o];
            C[o] = d;
            lsum += d * d;
          } else if (EPI == 2) {
            float g = softthr(eta * val, thr);
            Gbuf[o] = g;
            Zbuf[o] = g;
          } else if (EPI == 3) {
            float gold = Gbuf[o];
            float g = softthr(Zbuf[o] - eta * val, thr);
            Gbuf[o] = g;
            Zbuf[o] = g + mom * (g - gold);
          } else if (EPI == 4) {
            float t = val + Yb[cl];
            C[o] = t > 0.0f ? t : 0.0f;
          } else if (EPI == 5) {
            C[o] = val + Yb[cl];
          }
        }
      }
  if (EPI == 1) {
    red[tid] = lsum;
    __syncthreads();
    for (int s = 128; s > 0; s >>= 1) {
      if (tid < s) red[tid] += red[tid + s];
      __syncthreads();
    }
    if (tid == 0) atomicAdd(ssq, red[0]);
  }
}

// ---------------------------------------------------------------------------
// Power method on G = W^T W  (2048x2048): y = x @ G, plus per-block sum(y^2)
// ---------------------------------------------------------------------------
__global__ __launch_bounds__(256) void pm_matvec(const float* __restrict__ G,
                                                 const float* __restrict__ x,
                                                 float* __restrict__ y,
                                                 float* __restrict__ partial) {
  __shared__ float xs[256];
  __shared__ float red[256];
  const int j = blockIdx.x * 256 + threadIdx.x;  // grid = 8 blocks
  float s = 0.0f;
  for (int i0 = 0; i0 < 2048; i0 += 256) {
    xs[threadIdx.x] = x[i0 + threadIdx.x];
    __syncthreads();
#pragma unroll 8
    for (int i = 0; i < 256; ++i) s += xs[i] * G[(long long)(i0 + i) * 2048 + j];
    __syncthreads();
  }
  y[j] = s;
  red[threadIdx.x] = s * s;
  __syncthreads();
  for (int t = 128; t > 0; t >>= 1) {
    if (threadIdx.x < t) red[threadIdx.x] += red[threadIdx.x + t];
    __syncthreads();
  }
  if (threadIdx.x == 0) partial[blockIdx.x] = red[0];
}

__global__ __launch_bounds__(256) void pm_scale(const float* __restrict__ y,
                                                float* __restrict__ x,
                                                const float* __restrict__ partial,
                                                float* __restrict__ cout) {
  __shared__ float nmS;
  if (threadIdx.x == 0) {
    float s = 0.0f;
    for (int b = 0; b < 8; ++b) s += partial[b];
    nmS = sqrtf(s);
    cout[0] = nmS;  // last iteration's value = c = sigma_max(W)^2
  }
  __syncthreads();
  float inv = 1.0f / nmS;
  for (int j = threadIdx.x; j < 2048; j += 256) x[j] = y[j] * inv;
}

__global__ void init_x0(float* __restrict__ x) {
  int j = blockIdx.x * 256 + threadIdx.x;
  if (j < 2048) {
    unsigned h = (unsigned)j * 2654435761u ^ 0x9E3779B9u;
    h ^= h >> 16;
    h *= 0x85EBCA6Bu;
    h ^= h >> 13;
    x[j] = (float)(h & 0xFFFF) / 32768.0f - 1.0f;  // ~U(-1,1): fine for power iter
  }
}

__global__ __launch_bounds__(256) void sumsq_kernel(const float* __restrict__ p,
                                                    long long n,
                                                    float* __restrict__ out) {
  __shared__ float red[256];
  long long i = (long long)blockIdx.x * 256 + threadIdx.x;
  const long long stride = (long long)gridDim.x * 256;
  float s = 0.0f;
  for (; i < n; i += stride) {
    float v = p[i];
    s += v * v;
  }
  red[threadIdx.x] = s;
  __syncthreads();
  for (int t = 128; t > 0; t >>= 1) {
    if (threadIdx.x < t) red[threadIdx.x] += red[threadIdx.x + t];
    __syncthreads();
  }
  if (threadIdx.x == 0) atomicAdd(out, red[0]);
}

__global__ void zero_f32(float* __restrict__ p, int n) {
  int i = blockIdx.x * 256 + threadIdx.x;
  if (i < n) p[i] = 0.0f;
}

__global__ __launch_bounds__(64) void logsoftmax64(float* __restrict__ out) {
  __shared__ float red[64];
  float* p = out + (long long)blockIdx.x * 64;
  const int t = threadIdx.x;
  float v = p[t];
  red[t] = v;
  __syncthreads();
  for (int s = 32; s > 0; s >>= 1) {
    if (t < s) red[t] = fmaxf(red[t], red[t + s]);
    __syncthreads();
  }
  float mx = red[0];
  __syncthreads();
  red[t] = expf(v - mx);
  __syncthreads();
  for (int s = 32; s > 0; s >>= 1) {
    if (t < s) red[t] += red[t + s];
    __syncthreads();
  }
  float lse = logf(red[0]) + mx;
  p[t] = v - lse;
}

__global__ void finish_norms(const float* __restrict__ ssq,
                             const float* __restrict__ ssqY,
                             float* __restrict__ out) {
  int i = threadIdx.x;
  if (i < 20) out[i] = sqrtf(ssq[i] / ssqY[0]);  // ||R||/||Y||
}

// ---------------------------------------------------------------------------
extern "C" void kernel_launch(void* const* d_in, const int* in_sizes, int n_in,
                              void* d_out, int out_size, void* d_ws,
                              size_t ws_size, hipStream_t stream) {
  (void)in_sizes; (void)n_in; (void)out_size; (void)ws_size;

  const int NND = 2708, NF = 4096, MD = 2048, NH = 1024, NC = 64;

  const float* x     = (const float*)d_in[0];  // (2708,4096)
  const float* adj   = (const float*)d_in[1];  // (2708,2708)
  const float* gc1_w = (const float*)d_in[2];  // (4096,1024)
  const float* gc1_b = (const float*)d_in[3];  // (1024,)
  const float* gc2_w = (const float*)d_in[4];  // (1024,64)
  const float* gc2_b = (const float*)d_in[5];  // (64,)
  const float* W     = (const float*)d_in[6];  // (2708,2048)
  const int*   Kp    = (const int*)d_in[7];    // scalar lambda

  // outputs concatenated: log_softmax(2708,64), x_dec(2708,4096),
  //                       Gamma(2048,4096), norms(20,)
  float* out_ls    = (float*)d_out;
  float* out_xdec  = out_ls + (long long)NND * NC;
  float* Gamma     = out_xdec + (long long)NND * NF;
  float* out_norms = Gamma + (long long)MD * NF;

  char* wp = (char*)d_ws;
  auto alloc = [&](size_t bytes) -> float* {
    float* p = (float*)wp;
    wp += ((bytes + 255) / 256) * 256;
    return p;
  };
  float* Gm      = alloc((size_t)MD * MD * 4);    // W^T W
  float* Z       = alloc((size_t)MD * NF * 4);    // FISTA momentum iterate
  float* R       = alloc((size_t)NND * NF * 4);   // residual
  float* T1      = alloc((size_t)NND * NH * 4);   // x_dec @ gc1_w
  float* H       = alloc((size_t)NND * NH * 4);   // hidden
  float* T2      = alloc((size_t)NND * NC * 4);   // h @ gc2_w
  float* Xv      = alloc(2048 * 4);
  float* Yv      = alloc(2048 * 4);
  float* partial = alloc(16 * 4);
  float* cbuf    = alloc(4);
  float* ssqY    = alloc(4);
  float* ssq     = alloc(20 * 4);

  auto grid = [](int M, int N) { return dim3((N + BN - 1) / BN, (M + BM - 1) / BM); };

  // zero atomic accumulators (ws is not re-initialized between graph replays)
  zero_f32<<<1, 256, 0, stream>>>(ssqY, 1);
  zero_f32<<<1, 256, 0, stream>>>(ssq, 20);

  // ||Y||^2 over x
  sumsq_kernel<<<1024, 256, 0, stream>>>(x, (long long)NND * NF, ssqY);

  // G = W^T W   (2048 x 2048, K=2708)
  gemm_wmma<1, 0><<<grid(MD, MD), 256, 0, stream>>>(
      MD, MD, NND, W, MD, W, MD, Gm, MD, nullptr, nullptr, nullptr, nullptr,
      nullptr, 0.0f, nullptr);

  // power method: c = sigma_max(W)^2 via 100 iterations of x <- xG / ||xG||
  init_x0<<<8, 256, 0, stream>>>(Xv);
  for (int it = 0; it < 100; ++it) {
    pm_matvec<<<8, 256, 0, stream>>>(Gm, Xv, Yv, partial);
    pm_scale<<<1, 256, 0, stream>>>(Yv, Xv, partial, cbuf);
  }

  // Gamma0 = softthr(eta * W^T x, lam); Z = Gamma0   (2048 x 4096, K=2708)
  gemm_wmma<1, 2><<<grid(MD, NF), 256, 0, stream>>>(
      MD, NF, NND, W, MD, x, NF, nullptr, NF, nullptr, Gamma, Z, cbuf, Kp, 0.0f,
      nullptr);

  // FISTA iterations (momentum scalars are data-independent -> host side)
  float t = 1.0f;
  for (int it = 0; it < 20; ++it) {
    float tn = 0.5f * (1.0f + sqrtf(1.0f + 4.0f * t * t));
    float mom = (t - 1.0f) / tn;
    t = tn;
    // R = W @ Z - x ; accumulate ||R||^2 into ssq[it]
    gemm_wmma<0, 1><<<grid(NND, NF), 256, 0, stream>>>(
        NND, NF, MD, W, MD, Z, NF, R, NF, x, nullptr, nullptr, nullptr, nullptr,
        0.0f, ssq + it);
    // g = softthr(Z - eta * W^T R, lam/c); Z = g + mom*(g - Gamma); Gamma = g
    gemm_wmma<1, 3><<<grid(MD, NF), 256, 0, stream>>>(
        MD, NF, NND, W, MD, R, NF, nullptr, NF, nullptr, Gamma, Z, cbuf, Kp,
        mom, nullptr);
  }

  // x_dec = W @ Gamma   (2708 x 4096, K=2048)
  gemm_wmma<0, 0><<<grid(NND, NF), 256, 0, stream>>>(
      NND, NF, MD, W, MD, Gamma, NF, out_xdec, NF, nullptr, nullptr, nullptr,
      nullptr, nullptr, 0.0f, nullptr);

  // T1 = x_dec @ gc1_w   (2708 x 1024, K=4096)
  gemm_wmma<0, 0><<<grid(NND, NH), 256, 0, stream>>>(
      NND, NH, NF, out_xdec, NF, gc1_w, NH, T1, NH, nullptr, nullptr, nullptr,
      nullptr, nullptr, 0.0f, nullptr);

  // H = relu(adj @ T1 + b1)   (2708 x 1024, K=2708)
  gemm_wmma<0, 4><<<grid(NND, NH), 256, 0, stream>>>(
      NND, NH, NND, adj, NND, T1, NH, H, NH, gc1_b, nullptr, nullptr, nullptr,
      nullptr, 0.0f, nullptr);

  // T2 = H @ gc2_w   (2708 x 64, K=1024)
  gemm_wmma<0, 0><<<grid(NND, NC), 256, 0, stream>>>(
      NND, NC, NH, H, NH, gc2_w, NC, T2, NC, nullptr, nullptr, nullptr, nullptr,
      nullptr, 0.0f, nullptr);

  // OUT = adj @ T2 + b2   (2708 x 64, K=2708), then in-place log_softmax
  gemm_wmma<0, 5><<<grid(NND, NC), 256, 0, stream>>>(
      NND, NC, NND, adj, NND, T2, NC, out_ls, NC, gc2_b, nullptr, nullptr,
      nullptr, nullptr, 0.0f, nullptr);
  logsoftmax64<<<NND, 64, 0, stream>>>(out_ls);

  // norms[i] = ||R_i|| / ||Y||
  finish_norms<<<1, 32, 0, stream>>>(ssq, ssqY, out_norms);
}